// Scalar_65429531787740
// MI455X (gfx1250) — compile-verified
//
#include <hip/hip_runtime.h>
#include <stdint.h>

// ---------------- geometry constants (reference: 400x400 v, PML=20) ----------
#define DOMV   400        // velocity grid
#define PMLW   20
#define DOM    440        // padded computational domain (DOMV + 2*PML)
#define GUARD  4          // zero guard ring so stencil reads need no branches
#define P      448        // row pitch (floats): GUARD + DOM + GUARD
#define NROWS  448
#define GPLANE (448*448)  // floats per plane
#define TILE   32
#define NTILE  14         // ceil(440/32)
#define C1     (1.0f/48.0f)    // 1/(12*dh), dh=4
#define C2     (1.0f/192.0f)   // 1/(12*dh*dh)
#define DTC    0.0008f

// ---------------- gfx1250 async global->LDS copies (ASYNCcnt path) -----------
__device__ __forceinline__ void async_ld_b128(float* lds, const float* g) {
    // low 32 bits of a flat LDS pointer are the wave-relative LDS byte address
    unsigned l = (unsigned)(unsigned long long)lds;
    asm volatile("global_load_async_to_lds_b128 %0, %1, off"
                 :: "v"(l), "v"(g) : "memory");
}
__device__ __forceinline__ void wait_async0() {
    asm volatile("s_wait_asynccnt 0" ::: "memory");
}

// ---------------- init kernels ----------------------------------------------
__global__ void zero_kernel(float* p, size_t n) {
    size_t i = (size_t)blockIdx.x * blockDim.x + threadIdx.x;
    size_t s = (size_t)gridDim.x * blockDim.x;
    for (; i < n; i += s) p[i] = 0.0f;
}

__global__ void init_v2_kernel(const float* __restrict__ v, float* __restrict__ v2) {
    int i = blockIdx.x * blockDim.x + threadIdx.x;
    if (i >= DOM * DOM) return;
    int y = i / DOM, x = i - y * DOM;
    int iy = min(max(y - PMLW, 0), DOMV - 1);   // edge-replicate pad of v
    int ix = min(max(x - PMLW, 0), DOMV - 1);
    float val = v[iy * DOMV + ix];
    v2[(size_t)(y + GUARD) * P + (x + GUARD)] = val * val * (DTC * DTC);
}

__global__ void init_prof_kernel(float* ay, float* by, float* ax, float* bx) {
    int i = blockIdx.x * blockDim.x + threadIdx.x;
    if (i >= P) return;
    float a = 0.0f, b = 0.0f;          // guard entries: a=0,b=0 -> psi halo == 0
    int x = i - GUARD;
    if (x >= 0 && x < DOM) {
        const float pml = (float)PMLW;
        float sig_max = 3.0f * 2600.0f * logf(1000.0f) / (2.0f * pml * 4.0f);
        float dl = fminf(fmaxf((pml - (float)x) / pml, 0.0f), 1.0f);
        float dr = fminf(fmaxf(((float)x - (float)(DOM - 1) + pml) / pml, 0.0f), 1.0f);
        float d  = fmaxf(dl, dr);
        float sig = sig_max * d * d;
        b = expf(-sig * DTC);
        a = b - 1.0f;
    }
    ay[i] = a; by[i] = b; ax[i] = a; bx[i] = b;
}

// ---------------- fused timestep kernel -------------------------------------
// wfm_/wfp_ alias the same plane (leapfrog in place, per-point read-then-write).
// psi old/new are ping-ponged to avoid the inter-block halo race.
__global__ __launch_bounds__(256)
void step_kernel(const float* __restrict__ wfc_, const float* wfm_, float* wfp_,
                 const float* __restrict__ psiy_o_, const float* __restrict__ psix_o_,
                 float* __restrict__ psiy_n_, float* __restrict__ psix_n_,
                 float* __restrict__ zetay_, float* __restrict__ zetax_,
                 const float* __restrict__ v2, const float* __restrict__ pay,
                 const float* __restrict__ pby, const float* __restrict__ pax,
                 const float* __restrict__ pbx,
                 const int* __restrict__ src_loc, const float* __restrict__ src_amp,
                 const int* __restrict__ rec_loc, float* __restrict__ out,
                 int t, int nt, int nrec)
{
    const int b = blockIdx.z;
    const size_t bo = (size_t)b * GPLANE;
    const float* wfc    = wfc_    + bo;
    const float* wfm    = wfm_    + bo;
    float*       wfp    = wfp_    + bo;
    const float* psiy_o = psiy_o_ + bo;
    const float* psix_o = psix_o_ + bo;
    float*       psiy_n = psiy_n_ + bo;
    float*       psix_n = psix_n_ + bo;
    float*       zetay  = zetay_  + bo;
    float*       zetax  = zetax_  + bo;

    const int tid = threadIdx.y * 32 + threadIdx.x;
    const int x0 = blockIdx.x * TILE;
    const int y0 = blockIdx.y * TILE;

    __shared__ float s_w[40 * 40];    // wfc tile with halo 4
    __shared__ float s_py[36 * 32];   // psiy_new, y in [-2,34)
    __shared__ float s_px[32 * 36];   // psix_new, x in [-2,34)

    // Phase A: async-copy wfc (tile+halo) into LDS as 16B chunks.
    // Clamp rows to 447 and 4-float group starts to 444: everything beyond the
    // 440-domain lands in the guaranteed-zero guard ring == jnp.pad zero halo.
    for (int i = tid; i < 400; i += 256) {
        int ly = i / 10;
        int lg = (i - ly * 10) * 4;              // 0,4,...,36
        int py = min(y0 + ly, NROWS - 1);        // padded row = y0 - 4 + ly + GUARD
        int px = min(x0 + lg, P - 4);
        async_ld_b128(&s_w[ly * 40 + lg], wfc + (size_t)py * P + px);
    }

    // Overlap with the async fill: record receivers of the PREVIOUS step from
    // wfc (== previous wfp). These are plain vector loads, independent of LDS.
    if (blockIdx.x == 0 && blockIdx.y == 0 && t > 0) {
        for (int r = tid; r < nrec; r += 256) {
            int ry = rec_loc[(b * nrec + r) * 2 + 0] + PMLW + GUARD;
            int rx = rec_loc[(b * nrec + r) * 2 + 1] + PMLW + GUARD;
            out[((size_t)b * nrec + r) * nt + (t - 1)] = wfc[(size_t)ry * P + rx];
        }
    }

    wait_async0();
    __syncthreads();

    // Phase B: psi updates (recomputed in a +/-2 halo entirely from LDS wfc)
    for (int i = tid; i < 36 * 32; i += 256) {       // psiy_new
        int yl = i >> 5;                             // 0..35 -> y' = yl-2
        int xl = i & 31;
        int c = (yl + 2) * 40 + (xl + 4);            // wfc center for y'
        float d1 = (s_w[c - 80] - s_w[c + 80]
                    + 8.0f * (s_w[c + 40] - s_w[c - 40])) * C1;
        int prow = min(y0 + yl + 2, NROWS - 1);      // padded row of y'
        int pcol = min(x0 + xl + 4, NROWS - 1);
        float pold = psiy_o[(size_t)prow * P + pcol];
        s_py[i] = pby[prow] * pold + pay[prow] * d1;
    }
    for (int i = tid; i < 32 * 36; i += 256) {       // psix_new
        int yl = i / 36;
        int xr = i - yl * 36;                        // 0..35 -> x' = xr-2
        int c = (yl + 4) * 40 + (xr + 2);            // wfc center for x'
        float d1 = (s_w[c - 2] - s_w[c + 2]
                    + 8.0f * (s_w[c + 1] - s_w[c - 1])) * C1;
        int prow = min(y0 + yl + 4, NROWS - 1);
        int pcol = min(x0 + xr + 2, NROWS - 1);
        float pold = psix_o[(size_t)prow * P + pcol];
        s_px[i] = pbx[pcol] * pold + pax[pcol] * d1;
    }
    __syncthreads();

    // Source scalars (uniform per block -> scalar loads)
    int sy = src_loc[b * 2 + 0] + PMLW + GUARD;
    int sx = src_loc[b * 2 + 1] + PMLW + GUARD;
    float amp = src_amp[b * nt + t];

    // Phase C: second derivatives, zeta update, leapfrog, source, stores
    for (int k = 0; k < 4; ++k) {
        int yl = threadIdx.y + 8 * k;
        int xl = threadIdx.x;
        int gy = y0 + yl, gx = x0 + xl;
        if (gy >= DOM || gx >= DOM) continue;
        int prow = gy + GUARD, pcol = gx + GUARD;
        size_t pidx = (size_t)prow * P + pcol;

        int c = (yl + 4) * 40 + (xl + 4);
        float wc = s_w[c];
        float d2y = (-(s_w[c - 80] + s_w[c + 80])
                     + 16.0f * (s_w[c - 40] + s_w[c + 40]) - 30.0f * wc) * C2;
        float d2x = (-(s_w[c - 2] + s_w[c + 2])
                     + 16.0f * (s_w[c - 1] + s_w[c + 1]) - 30.0f * wc) * C2;

        int py_c = (yl + 2) * 32 + xl;               // psiy_new at (gy,gx)
        d2y += (s_py[py_c - 64] - s_py[py_c + 64]
                + 8.0f * (s_py[py_c + 32] - s_py[py_c - 32])) * C1;
        int px_c = yl * 36 + (xl + 2);               // psix_new at (gy,gx)
        d2x += (s_px[px_c - 2] - s_px[px_c + 2]
                + 8.0f * (s_px[px_c + 1] - s_px[px_c - 1])) * C1;

        float zy = pby[prow] * zetay[pidx] + pay[prow] * d2y;
        float zx = pbx[pcol] * zetax[pidx] + pax[pcol] * d2x;
        float v2c = v2[pidx];
        float wp = v2c * (d2y + zy + d2x + zx) + 2.0f * wc - wfm[pidx];
        if (prow == sy && pcol == sx) wp += v2c * amp;

        wfp[pidx]    = wp;
        zetay[pidx]  = zy;
        zetax[pidx]  = zx;
        psiy_n[pidx] = s_py[py_c];
        psix_n[pidx] = s_px[px_c];
    }
}

__global__ void record_kernel(const float* __restrict__ wf, const int* __restrict__ rec,
                              float* __restrict__ out, int nrec, int nt, int t) {
    int b = blockIdx.x;
    const float* w = wf + (size_t)b * GPLANE;
    for (int r = threadIdx.x; r < nrec; r += blockDim.x) {
        int ry = rec[(b * nrec + r) * 2 + 0] + PMLW + GUARD;
        int rx = rec[(b * nrec + r) * 2 + 1] + PMLW + GUARD;
        out[((size_t)b * nrec + r) * nt + t] = w[(size_t)ry * P + rx];
    }
}

// ---------------- host orchestration ----------------------------------------
extern "C" void kernel_launch(void* const* d_in, const int* in_sizes, int n_in,
                              void* d_out, int out_size, void* d_ws, size_t ws_size,
                              hipStream_t stream) {
    const float* v    = (const float*)d_in[0];
    const float* amp  = (const float*)d_in[1];   // (B, 1, nt)
    const int*   sloc = (const int*)d_in[2];     // (B, 1, 2)
    const int*   rloc = (const int*)d_in[3];     // (B, nrec, 2)
    float*       out  = (float*)d_out;           // (B, nrec, nt)

    const int B    = in_sizes[2] / 2;            // n_src == 1
    const int nt   = in_sizes[1] / B;
    const int nrec = in_sizes[3] / (2 * B);

    const size_t g = (size_t)GPLANE;
    float* W     = (float*)d_ws;
    float* v2    = W;                            // 1 plane (shared across batch)
    float* ay    = W + g;
    float* by    = ay + P;
    float* ax    = by + P;
    float* bx    = ax + P;
    float* base  = W + g + 4 * (size_t)P;        // 8 field planes per batch
    float* wfA   = base;
    float* wfB   = wfA   + (size_t)B * g;
    float* psiyA = wfB   + (size_t)B * g;
    float* psiyB = psiyA + (size_t)B * g;
    float* psixA = psiyB + (size_t)B * g;
    float* psixB = psixA + (size_t)B * g;
    float* zetay = psixB + (size_t)B * g;
    float* zetax = zetay + (size_t)B * g;

    // Zero entire used workspace (fields must start at 0; guards stay 0 forever)
    size_t total = g + 4 * (size_t)P + (size_t)8 * B * g;
    zero_kernel<<<1024, 256, 0, stream>>>(W, total);
    init_v2_kernel<<<(DOM * DOM + 255) / 256, 256, 0, stream>>>(v, v2);
    init_prof_kernel<<<2, 256, 0, stream>>>(ay, by, ax, bx);

    dim3 grid(NTILE, NTILE, B);
    dim3 blk(32, 8);
    for (int t = 0; t < nt; ++t) {
        const bool even = (t & 1) == 0;
        const float* wfc = even ? wfA : wfB;
        float*       wfn = even ? wfB : wfA;     // wfm and wfp (in place)
        const float* pyo = even ? psiyA : psiyB;
        float*       pyn = even ? psiyB : psiyA;
        const float* pxo = even ? psixA : psixB;
        float*       pxn = even ? psixB : psixA;
        step_kernel<<<grid, blk, 0, stream>>>(wfc, wfn, wfn, pyo, pxo, pyn, pxn,
                                              zetay, zetax, v2, ay, by, ax, bx,
                                              sloc, amp, rloc, out, t, nt, nrec);
    }
    const float* wfin = ((nt & 1) == 0) ? wfA : wfB;  // wfp of last step
    record_kernel<<<B, 128, 0, stream>>>(wfin, rloc, out, nrec, nt, nt - 1);
}